// gu_HiPPO_LSI_46574625358287
// MI455X (gfx1250) — compile-verified
//
#include <hip/hip_runtime.h>

// HiPPO-LegS scan on gfx1250:  c_t = A_t @ c_{t-1} + B_t * x_t  (N=64)
// f32 WMMA (16x16x4) for full-precision 1024-step recurrence.
// Optimizations vs v1:
//  - 4 independent WMMA accumulator chains (depth ~5 instead of 17) since the
//    scan leaves only ~1-2 waves/SIMD to hide XDL latency.
//  - register double-buffering of A_t fragments + x_t/B_t scalars: step t+1's
//    global loads issue before step t's WMMA chain, hiding L2 latency behind
//    compute + barrier.

typedef __attribute__((ext_vector_type(2))) float v2f;
typedef __attribute__((ext_vector_type(8))) float v8f;

#define WMMA_F32(Af, Bf, Cacc) \
    __builtin_amdgcn_wmma_f32_16x16x4_f32(false, (Af), false, (Bf), (short)0, (Cacc), false, false)

// One scan step: 16 K-chunks over 4 accumulator chains + rank-1 input WMMA,
// NT-store result, hand new state to all waves via LDS ping-pong.
#define STEP_BODY(T, F, XV, BVv) do {                                          \
    v8f a0 = {}, a1 = {}, a2 = {}, a3 = {};                                    \
    const float* crow = &cbuf[cur][l16 * 64 + 2 * lhalf];                      \
    _Pragma("unroll")                                                          \
    for (int kc = 0; kc < 16; kc += 4) {                                       \
        v2f c0 = *(const v2f*)(crow + (kc + 0) * 4);                           \
        v2f c1 = *(const v2f*)(crow + (kc + 1) * 4);                           \
        v2f c2 = *(const v2f*)(crow + (kc + 2) * 4);                           \
        v2f c3 = *(const v2f*)(crow + (kc + 3) * 4);                           \
        a0 = WMMA_F32(c0, F[kc + 0], a0);                                      \
        a1 = WMMA_F32(c1, F[kc + 1], a1);                                      \
        a2 = WMMA_F32(c2, F[kc + 2], a2);                                      \
        a3 = WMMA_F32(c3, F[kc + 3], a3);                                      \
    }                                                                          \
    v2f ae, be;                                                                \
    ae.x = lhalf ? 0.0f : (XV);  ae.y = 0.0f;                                  \
    be.x = lhalf ? 0.0f : (BVv); be.y = 0.0f;                                  \
    a1 = WMMA_F32(ae, be, a1);   /* rank-1 input injection, chain 1 */         \
    v8f s = (a0 + a1) + (a2 + a3);                                             \
    const float* ap = (const float*)&s;                                        \
    float* obase = out + ((size_t)(T) * batch + bbase + 8 * lhalf) * 64 + n;   \
    float* cnew  = &cbuf[cur ^ 1][(8 * lhalf) * 64 + n];                       \
    _Pragma("unroll")                                                          \
    for (int r = 0; r < 8; ++r) {                                              \
        __builtin_nontemporal_store(ap[r], obase + (size_t)r * 64);            \
        cnew[r * 64] = ap[r];                                                  \
    }                                                                          \
    __syncthreads();                                                           \
    cur ^= 1;                                                                  \
} while (0)

__global__ __launch_bounds__(128) void gu_hippo_scan_wmma(
    const float* __restrict__ x,    // (L, batch)
    const float* __restrict__ A,    // (L, 64, 64) row-major (n,k)
    const float* __restrict__ Bv,   // (L, 64)
    float* __restrict__ out,        // (L, batch, 64)
    int L, int batch)
{
    __shared__ float cbuf[2][16 * 64];   // ping-pong state: 16 batch rows x 64 dims

    const int tid   = threadIdx.x;
    const int wave  = tid >> 5;
    const int lane  = tid & 31;
    const int l16   = lane & 15;
    const int lhalf = lane >> 4;
    const int bbase = blockIdx.x * 16;
    const int n     = wave * 16 + l16;

    for (int i = tid; i < 16 * 64; i += 128) cbuf[0][i] = 0.0f;
    __syncthreads();

    const size_t arow = (size_t)n * 64 + 2 * lhalf;  // lane's A_t^T fragment base

    v2f fA[16], fB[16];          // double-buffered B-operand fragments of A_t
    float xA, bA, xB, bB;        // double-buffered x_t / B_t scalars

    // prologue: fragments for t = 0
    {
        #pragma unroll
        for (int kc = 0; kc < 16; ++kc)
            fA[kc] = *(const v2f*)(A + arow + kc * 4);
        xA = x[bbase + l16];
        bA = Bv[n];
    }

    int cur = 0;
    for (int t = 0; t < L; t += 2) {
        // even step: prefetch t+1 into fB, compute t with fA
        {
            const int tn = (t + 1 < L) ? (t + 1) : t;
            const float* Atn = A + (size_t)tn * 4096;
            #pragma unroll
            for (int kc = 0; kc < 16; ++kc)
                fB[kc] = *(const v2f*)(Atn + arow + kc * 4);
            xB = x[(size_t)tn * batch + bbase + l16];
            bB = Bv[(size_t)tn * 64 + n];

            STEP_BODY(t, fA, xA, bA);
        }
        // odd step: prefetch t+2 into fA, compute t+1 with fB
        if (t + 1 < L) {
            const int tn = (t + 2 < L) ? (t + 2) : (t + 1);
            const float* Atn = A + (size_t)tn * 4096;
            #pragma unroll
            for (int kc = 0; kc < 16; ++kc)
                fA[kc] = *(const v2f*)(Atn + arow + kc * 4);
            xA = x[(size_t)tn * batch + bbase + l16];
            bA = Bv[(size_t)tn * 64 + n];

            STEP_BODY(t + 1, fB, xB, bB);
        }
    }
}

extern "C" void kernel_launch(void* const* d_in, const int* in_sizes, int n_in,
                              void* d_out, int out_size, void* d_ws, size_t ws_size,
                              hipStream_t stream) {
    const float* x  = (const float*)d_in[0];   // inputs (L, batch)
    const float* A  = (const float*)d_in[1];   // A_stacked (L, 64, 64)
    const float* Bv = (const float*)d_in[2];   // B_stacked (L, 64)
    float* out = (float*)d_out;

    const int L      = in_sizes[2] / 64;
    const int batch  = in_sizes[0] / L;
    const int blocks = batch / 16;             // 16 batch rows per 4-wave block

    gu_hippo_scan_wmma<<<blocks, 128, 0, stream>>>(x, A, Bv, out, L, batch);
}